// EGAT_70179765616704
// MI455X (gfx1250) — compile-verified
//
#include <hip/hip_runtime.h>
#include <stdint.h>

typedef __attribute__((ext_vector_type(2))) float v2f;
typedef __attribute__((ext_vector_type(8))) float v8f;

__device__ __forceinline__ float lrelu(float v) { return v > 0.f ? v : 0.2f * v; }

__device__ __forceinline__ v8f wmma4(v2f a, v2f b, v8f c) {
    return __builtin_amdgcn_wmma_f32_16x16x4_f32(false, a, false, b, (short)0, c, false, false);
}

// --- CDNA5 async memory->LDS path (ASYNCcnt), VGLOBAL encoding, GV mode ----
__device__ __forceinline__ uint32_t lds_off32(const void* p) {
    // generic LDS address = {shared_aperture[63:32], lds_offset[31:0]} (ISA 10.2)
    return (uint32_t)(uintptr_t)p;
}
__device__ __forceinline__ void async_copy_b128(uint32_t lds_off, const float* gptr) {
    asm volatile("global_load_async_to_lds_b128 %0, %1, off"
                 :
                 : "v"(lds_off), "v"((uint64_t)(uintptr_t)gptr)
                 : "memory");
}
__device__ __forceinline__ void wait_async0() {
    asm volatile("s_wait_asynccnt 0x0" ::: "memory");
}

// ---------------------------------------------------------------------------
// Stage 1: A[i,:] = softmax_j( exp(-sqrt(max(||x_i-x_j||^2, 1e-12))) )
// ---------------------------------------------------------------------------
__global__ void k_rowA(const float* __restrict__ x, float* __restrict__ A, int n) {
    __shared__ float row[4096];
    __shared__ float sred[256];
    const int i = blockIdx.x;
    const int t = threadIdx.x;
    const float xi0 = x[2 * i], xi1 = x[2 * i + 1];

    float mymax = -1e30f;
    for (int j = t; j < n; j += 256) {
        float d0 = xi0 - x[2 * j];
        float d1 = xi1 - x[2 * j + 1];
        float d2 = fmaxf(d0 * d0 + d1 * d1, 1e-12f);
        float D = __expf(-sqrtf(d2));
        row[j] = D;
        mymax = fmaxf(mymax, D);
    }
    sred[t] = mymax;
    __syncthreads();
    for (int s = 128; s > 0; s >>= 1) {
        if (t < s) sred[t] = fmaxf(sred[t], sred[t + s]);
        __syncthreads();
    }
    const float m = sred[0];
    __syncthreads();

    float mysum = 0.f;
    for (int j = t; j < n; j += 256) {
        float e = __expf(row[j] - m);
        row[j] = e;
        mysum += e;
    }
    sred[t] = mysum;
    __syncthreads();
    for (int s = 128; s > 0; s >>= 1) {
        if (t < s) sred[t] += sred[t + s];
        __syncthreads();
    }
    const float inv = 1.0f / sred[0];
    for (int j = t; j < n; j += 256) A[(size_t)i * n + j] = row[j] * inv;
}

// ---------------------------------------------------------------------------
// Stage 2: invc[j] = 1 / sum_i A[i,j]
// ---------------------------------------------------------------------------
__global__ void k_colsum_inv(const float* __restrict__ A, float* __restrict__ invc, int n) {
    const int j = blockIdx.x * 256 + threadIdx.x;
    float s = 0.f;
    for (int i = 0; i < n; ++i) s += A[(size_t)i * n + j];
    invc[j] = 1.0f / s;
}

// ---------------------------------------------------------------------------
// Stage 2b: As = A * colscale (columns), so the NT GEMM can use raw async
// tile copies (the async engine writes LDS directly; no fused scaling).
// ---------------------------------------------------------------------------
__global__ void k_colscale(const float* __restrict__ A, const float* __restrict__ invc,
                           float* __restrict__ As, int n) {
    size_t gid = (size_t)blockIdx.x * 256 + threadIdx.x;
    size_t base = gid * 4;
    int j = (int)(base & (size_t)(n - 1));  // n is a power of two
    float4 a = *(const float4*)(A + base);
    float4 s = *(const float4*)(invc + j);
    float4 r;
    r.x = a.x * s.x; r.y = a.y * s.y; r.z = a.z * s.z; r.w = a.w * s.w;
    *(float4*)(As + base) = r;
}

// ---------------------------------------------------------------------------
// Stage 3: Af = As @ A^T  (M=N=K=n, f32 WMMA 16x16x4)
// Double-buffered LDS fed by global_load_async_to_lds_b128 (ASYNCcnt).
// Block 256 thr -> 128x64 tile; 8 waves as 4x2, each wave 32x32 (2x2 WMMA).
// ---------------------------------------------------------------------------
#define PAD_A 36  // 36 floats = 144B row stride: 16B aligned, conflict-free

__global__ void k_gemm_nt_async(const float* __restrict__ As, const float* __restrict__ A,
                                float* __restrict__ C, int n) {
    __shared__ float At[2][128 * PAD_A];
    __shared__ float Bt[2][64 * PAD_A];
    const int i0 = blockIdx.x * 128;
    const int j0 = blockIdx.y * 64;
    const int t = threadIdx.x;
    const int lane = t & 31, w = t >> 5;
    const int m0w = (w & 3) * 32, n0w = (w >> 2) * 32;
    const int lm = lane & 15, half = lane >> 4;

    // Per-thread async copy assignment (16B per issue)
    const int ra = t >> 3;          // 0..31
    const int ca = (t & 7) * 4;     // 0..28

    v8f acc[2][2] = {};

    auto issue = [&](int buf, int k0) {
        // A tile: 128x32 floats, 4 x b128 per thread
        for (int p = 0; p < 4; ++p) {
            int r = p * 32 + ra;
            async_copy_b128(lds_off32(&At[buf][r * PAD_A + ca]),
                            As + (size_t)(i0 + r) * n + k0 + ca);
        }
        // B tile: 64x32 floats, 2 x b128 per thread
        for (int p = 0; p < 2; ++p) {
            int r = p * 32 + ra;
            async_copy_b128(lds_off32(&Bt[buf][r * PAD_A + ca]),
                            A + (size_t)(j0 + r) * n + k0 + ca);
        }
    };

    issue(0, 0);
    wait_async0();
    __syncthreads();

    for (int k0 = 0; k0 < n; k0 += 32) {
        const int buf = (k0 >> 5) & 1;
        if (k0 + 32 < n) issue(buf ^ 1, k0 + 32);  // prefetch next chunk

        const float* Ab = &At[buf][0];
        const float* Bb = &Bt[buf][0];
        for (int kk = 0; kk < 32; kk += 4) {
            v2f a[2], b[2];
            for (int mi = 0; mi < 2; ++mi) {
                const float* pA = Ab + (m0w + mi * 16 + lm) * PAD_A + kk + 2 * half;
                a[mi].x = pA[0];
                a[mi].y = pA[1];
            }
            for (int ni = 0; ni < 2; ++ni) {
                const float* pB = Bb + (n0w + ni * 16 + lm) * PAD_A + kk + 2 * half;
                b[ni].x = pB[0];
                b[ni].y = pB[1];
            }
            acc[0][0] = wmma4(a[0], b[0], acc[0][0]);
            acc[0][1] = wmma4(a[0], b[1], acc[0][1]);
            acc[1][0] = wmma4(a[1], b[0], acc[1][0]);
            acc[1][1] = wmma4(a[1], b[1], acc[1][1]);
        }

        wait_async0();   // own prefetch landed
        __syncthreads(); // everyone done reading buf; prefetch visible to all
    }

    for (int mi = 0; mi < 2; ++mi)
        for (int ni = 0; ni < 2; ++ni)
            for (int r = 0; r < 8; ++r) {
                size_t row = (size_t)(i0 + m0w + mi * 16 + r + 8 * half);
                C[row * n + j0 + n0w + ni * 16 + lm] = acc[mi][ni][r];
            }
}

// ---------------------------------------------------------------------------
// Generic NN WMMA GEMM: C[M,N] = A[M,K] @ B[K,N] (+bias, +leaky_relu)
// Requires M%128==0, K%32==0. N guarded (supports N=32 or 64).
// (kept synchronous: guarded B-tile loads would be OOB under async copies)
// ---------------------------------------------------------------------------
#define PAD_B 68

__global__ void k_gemm_nn(const float* __restrict__ A, const float* __restrict__ B,
                          const float* __restrict__ bias, float* __restrict__ C,
                          int M, int N, int K, int applyLrelu) {
    __shared__ float At[128 * PAD_A];
    __shared__ float Bt[32 * PAD_B];
    const int i0 = blockIdx.x * 128;
    const int j0 = blockIdx.y * 64;
    const int t = threadIdx.x;
    const int lane = t & 31, w = t >> 5;
    const int m0w = (w & 3) * 32, n0w = (w >> 2) * 32;
    const int lm = lane & 15, half = lane >> 4;

    v8f acc[2][2] = {};

    for (int k0 = 0; k0 < K; k0 += 32) {
        for (int p = 0; p < 4; ++p) {
            int r = p * 32 + (t >> 3);
            int c = (t & 7) * 4;
            float4 v = *(const float4*)(A + (size_t)(i0 + r) * K + k0 + c);
            At[r * PAD_A + c + 0] = v.x;
            At[r * PAD_A + c + 1] = v.y;
            At[r * PAD_A + c + 2] = v.z;
            At[r * PAD_A + c + 3] = v.w;
        }
        for (int p = 0; p < 2; ++p) {
            int r = p * 16 + (t >> 4);
            int c = (t & 15) * 4;
            for (int q = 0; q < 4; ++q) {
                int gc = j0 + c + q;
                Bt[r * PAD_B + c + q] = (gc < N) ? B[(size_t)(k0 + r) * N + gc] : 0.f;
            }
        }
        __syncthreads();

        for (int kk = 0; kk < 32; kk += 4) {
            v2f a[2], b[2];
            for (int mi = 0; mi < 2; ++mi) {
                const float* pA = &At[(m0w + mi * 16 + lm) * PAD_A + kk + 2 * half];
                a[mi].x = pA[0];
                a[mi].y = pA[1];
            }
            for (int ni = 0; ni < 2; ++ni) {
                int col = n0w + ni * 16 + lm;
                b[ni].x = Bt[(kk + 2 * half) * PAD_B + col];
                b[ni].y = Bt[(kk + 2 * half + 1) * PAD_B + col];
            }
            acc[0][0] = wmma4(a[0], b[0], acc[0][0]);
            acc[0][1] = wmma4(a[0], b[1], acc[0][1]);
            acc[1][0] = wmma4(a[1], b[0], acc[1][0]);
            acc[1][1] = wmma4(a[1], b[1], acc[1][1]);
        }
        __syncthreads();
    }

    for (int mi = 0; mi < 2; ++mi)
        for (int ni = 0; ni < 2; ++ni) {
            int col = j0 + n0w + ni * 16 + lm;
            if (col < N) {
                for (int r = 0; r < 8; ++r) {
                    float v = acc[mi][ni][r];
                    if (bias) v += bias[col];
                    if (applyLrelu) v = lrelu(v);
                    C[(size_t)(i0 + m0w + mi * 16 + r + 8 * half) * N + col] = v;
                }
            }
        }
}

// ---------------------------------------------------------------------------
// s = x @ Wfc  (K=2, trivial)
// ---------------------------------------------------------------------------
__global__ void k_s(const float* __restrict__ x, const float* __restrict__ Wfc,
                    float* __restrict__ s, int n) {
    int gid = blockIdx.x * 256 + threadIdx.x;
    int i = gid >> 6, c = gid & 63;
    if (i < n) s[(size_t)i * 64 + c] = x[2 * i] * Wfc[c] + x[2 * i + 1] * Wfc[64 + c];
}

// ---------------------------------------------------------------------------
// src[h,i] = hp[h,i,:].asrc[h]; dst[h,i] = hp[h,i,:].adst[h]   (f_out = 32)
// ---------------------------------------------------------------------------
__global__ void k_srcdst(const float* __restrict__ hp, const float* __restrict__ asrc,
                         const float* __restrict__ adst, float* __restrict__ src,
                         float* __restrict__ dst, int n) {
    int gid = blockIdx.x * 256 + threadIdx.x;
    int h = gid / n, i = gid % n;
    const float* r = hp + ((size_t)h * n + i) * 32;
    float s = 0.f, d = 0.f;
    for (int c = 0; c < 32; ++c) {
        s += r[c] * asrc[h * 32 + c];
        d += r[c] * adst[h * 32 + c];
    }
    src[(size_t)h * n + i] = s;
    dst[(size_t)h * n + i] = d;
}

__global__ void k_dstmax(const float* __restrict__ dst, float* __restrict__ dmax, int n) {
    __shared__ float sred[256];
    int h = blockIdx.x, t = threadIdx.x;
    float m = -1e30f;
    for (int j = t; j < n; j += 256) m = fmaxf(m, dst[(size_t)h * n + j]);
    sred[t] = m;
    __syncthreads();
    for (int s = 128; s > 0; s >>= 1) {
        if (t < s) sred[t] = fmaxf(sred[t], sred[t + s]);
        __syncthreads();
    }
    if (t == 0) dmax[h] = sred[0];
}

// ---------------------------------------------------------------------------
// Fused attention: out[i, h*32+c] = sum_j softmax_j(lrelu(src_i+dst_j)) hp[h,j,c] + b[c]
// One wave per (head, 16-row tile); probabilities generated on the fly as the
// WMMA A-fragment (flash-attention style, nothing materialized).
// ---------------------------------------------------------------------------
__global__ void k_attn(const float* __restrict__ hp, const float* __restrict__ src,
                       const float* __restrict__ dst, const float* __restrict__ dmax,
                       const float* __restrict__ bias, float* __restrict__ out,
                       int n, int H) {
    const int h = blockIdx.y;
    const int w = threadIdx.x >> 5;
    const int lane = threadIdx.x & 31;
    const int i0 = (blockIdx.x * 4 + w) * 16;
    const int lm = lane & 15, half = lane >> 4;

    const float* srcH = src + (size_t)h * n;
    const float* dstH = dst + (size_t)h * n;
    const float* hpH = hp + (size_t)h * n * 32;

    const float si = srcH[i0 + lm];
    const float m = lrelu(si + dmax[h]);  // lrelu monotone -> exact row max

    float sum = 0.f;
    const int jb = half * (n >> 1), je = jb + (n >> 1);
    for (int j = jb; j < je; ++j) sum += __expf(lrelu(si + dstH[j]) - m);
    sum += __shfl_xor(sum, 16, 32);
    const float inv = 1.0f / sum;

    v8f acc[2] = {};
    for (int kk = 0; kk < n; kk += 4) {
        const int j = kk + 2 * half;
        v2f a;
        a.x = __expf(lrelu(si + dstH[j]) - m) * inv;
        a.y = __expf(lrelu(si + dstH[j + 1]) - m) * inv;
        v2f b0, b1;
        b0.x = hpH[(size_t)j * 32 + lm];
        b0.y = hpH[(size_t)(j + 1) * 32 + lm];
        b1.x = hpH[(size_t)j * 32 + 16 + lm];
        b1.y = hpH[(size_t)(j + 1) * 32 + 16 + lm];
        acc[0] = wmma4(a, b0, acc[0]);
        acc[1] = wmma4(a, b1, acc[1]);
    }

    const int stride = H * 32;
    for (int ni = 0; ni < 2; ++ni) {
        int col = ni * 16 + lm;
        for (int r = 0; r < 8; ++r)
            out[(size_t)(i0 + r + 8 * half) * stride + h * 32 + col] = acc[ni][r] + bias[col];
    }
}

// ---------------------------------------------------------------------------
extern "C" void kernel_launch(void* const* d_in, const int* in_sizes, int n_in,
                              void* d_out, int out_size, void* d_ws, size_t ws_size,
                              hipStream_t stream) {
    (void)in_sizes; (void)n_in; (void)out_size; (void)ws_size;
    const int n = 4096, H = 4;
    const float* x     = (const float*)d_in[0];
    const float* Wfc   = (const float*)d_in[1];
    const float* b_gn  = (const float*)d_in[2];
    const float* w0    = (const float*)d_in[3];
    const float* asrc0 = (const float*)d_in[4];
    const float* adst0 = (const float*)d_in[5];
    const float* b0    = (const float*)d_in[6];
    const float* w1    = (const float*)d_in[7];
    const float* asrc1 = (const float*)d_in[8];
    const float* adst1 = (const float*)d_in[9];
    const float* b1    = (const float*)d_in[10];
    float* out = (float*)d_out;

    char* p = (char*)d_ws;
    auto alloc = [&](size_t elems) {
        float* r = (float*)p;
        p += ((elems * sizeof(float)) + 255) & ~(size_t)255;
        return r;
    };
    float* A     = alloc((size_t)n * n);      // 64 MB
    float* As    = alloc((size_t)n * n);      // 64 MB (col-scaled A)
    float* Af    = alloc((size_t)n * n);      // 64 MB
    float* invc  = alloc(n);
    float* s     = alloc((size_t)n * 64);
    float* h0    = alloc((size_t)n * 64);
    float* hp0   = alloc((size_t)H * n * 32);
    float* src0  = alloc((size_t)H * n);
    float* dst0  = alloc((size_t)H * n);
    float* dmax0 = alloc(H);
    float* h1    = alloc((size_t)n * 128);
    float* hp1   = alloc((size_t)H * n * 32);
    float* src1  = alloc((size_t)H * n);
    float* dst1  = alloc((size_t)H * n);
    float* dmax1 = alloc(H);

    // Graph construction
    k_rowA<<<n, 256, 0, stream>>>(x, A, n);
    k_colsum_inv<<<n / 256, 256, 0, stream>>>(A, invc, n);
    k_colscale<<<(int)(((size_t)n * n / 4) / 256), 256, 0, stream>>>(A, invc, As, n);
    k_gemm_nt_async<<<dim3(n / 128, n / 64), 256, 0, stream>>>(As, A, Af, n);
    k_s<<<(n * 64) / 256, 256, 0, stream>>>(x, Wfc, s, n);
    k_gemm_nn<<<dim3(n / 128, 1), 256, 0, stream>>>(Af, s, b_gn, h0, n, 64, n, 1);

    // MHGA layer 0 (f_in=64, f_out=32, H=4) -> h1 [n, 128]
    for (int h = 0; h < H; ++h)
        k_gemm_nn<<<dim3(n / 128, 1), 256, 0, stream>>>(
            h0, w0 + (size_t)h * 64 * 32, nullptr, hp0 + (size_t)h * n * 32, n, 32, 64, 0);
    k_srcdst<<<(H * n) / 256, 256, 0, stream>>>(hp0, asrc0, adst0, src0, dst0, n);
    k_dstmax<<<H, 256, 0, stream>>>(dst0, dmax0, n);
    k_attn<<<dim3(n / 64, H), 128, 0, stream>>>(hp0, src0, dst0, dmax0, b0, h1, n, H);

    // MHGA layer 1 (f_in=128, f_out=32, H=4) -> out [n, 128]
    for (int h = 0; h < H; ++h)
        k_gemm_nn<<<dim3(n / 128, 1), 256, 0, stream>>>(
            h1, w1 + (size_t)h * 128 * 32, nullptr, hp1 + (size_t)h * n * 32, n, 32, 128, 0);
    k_srcdst<<<(H * n) / 256, 256, 0, stream>>>(hp1, asrc1, adst1, src1, dst1, n);
    k_dstmax<<<H, 256, 0, stream>>>(dst1, dmax1, n);
    k_attn<<<dim3(n / 64, H), 128, 0, stream>>>(hp1, src1, dst1, dmax1, b1, out, n, H);
}